// LightGCN_54022098649194
// MI455X (gfx1250) — compile-verified
//
#include <hip/hip_runtime.h>

typedef float v2f __attribute__((ext_vector_type(2)));
typedef float v8f __attribute__((ext_vector_type(8)));

#define NUSERS  100000
#define NITEMS  50000
#define NTOT    150000
#define EMB_D   64
#define NNZ_E   4800000
#define BATCH_B 16384
#define NLAYER  3

// Native fire-and-forget f32 atomic add at device scope (RMW executes at L2;
// no return value -> STOREcnt-tracked, no VGPR writeback, no CAS loop).
__device__ __forceinline__ void atom_add_f32_dev(float* p, float v) {
  asm volatile("global_atomic_add_f32 %0, %1, off scope:SCOPE_DEV"
               :
               : "v"(p), "v"(v)
               : "memory");
}

// ---------------------------------------------------------------------------
// acc = emb = concat(user_emb, item_emb), float4 granularity.
// ---------------------------------------------------------------------------
__global__ void lg_init(const float* __restrict__ ue, const float* __restrict__ ie,
                        float* __restrict__ acc, float* __restrict__ emb) {
  long i = (long)blockIdx.x * blockDim.x + threadIdx.x;  // float4 index
  const long nU4 = (long)NUSERS * EMB_D / 4;
  const long nT4 = (long)NTOT * EMB_D / 4;
  if (i >= nT4) return;
  float4 v = (i < nU4) ? ((const float4*)ue)[i] : ((const float4*)ie)[i - nU4];
  ((float4*)acc)[i] = v;
  ((float4*)emb)[i] = v;
}

__global__ void lg_zero(float* __restrict__ p) {
  long i = (long)blockIdx.x * blockDim.x + threadIdx.x;
  const long nT4 = (long)NTOT * EMB_D / 4;
  if (i < nT4) ((float4*)p)[i] = make_float4(0.f, 0.f, 0.f, 0.f);
}

// ---------------------------------------------------------------------------
// Edge-parallel SpMM scatter: 16 threads per edge, float4 per thread (=64 dims).
// eout[r][:] += vals[e] * ein[c][:] via native f32 L2 atomics (table is
// L2-resident: 38.4 MB << 192 MB, so both the gather and the atomics hit L2).
// ---------------------------------------------------------------------------
__global__ void lg_spmm(const int* __restrict__ rows, const int* __restrict__ cols,
                        const float* __restrict__ vals,
                        const float* __restrict__ ein, float* __restrict__ eout) {
  long gid = (long)blockIdx.x * blockDim.x + threadIdx.x;
  long e = gid >> 4;
  if (e >= NNZ_E) return;
  int d4 = (int)(gid & 15) << 2;

  if ((gid & 15) == 0) {
    // keep the streaming edge arrays ahead of us in cache
    __builtin_prefetch(rows + e + 2048, 0, 0);
    __builtin_prefetch(cols + e + 2048, 0, 0);
    __builtin_prefetch(vals + e + 2048, 0, 0);
  }

  int   r = rows[e];
  int   c = cols[e];
  float v = vals[e];

  float4 x = *(const float4*)(ein + (long)c * EMB_D + d4);
  float* dst = eout + (long)r * EMB_D + d4;
  atom_add_f32_dev(dst + 0, v * x.x);
  atom_add_f32_dev(dst + 1, v * x.y);
  atom_add_f32_dev(dst + 2, v * x.z);
  atom_add_f32_dev(dst + 3, v * x.w);
}

__global__ void lg_accum(float* __restrict__ acc, const float* __restrict__ e) {
  long i = (long)blockIdx.x * blockDim.x + threadIdx.x;
  const long nT4 = (long)NTOT * EMB_D / 4;
  if (i >= nT4) return;
  float4 a = ((const float4*)acc)[i];
  float4 b = ((const float4*)e)[i];
  a.x += b.x; a.y += b.y; a.z += b.z; a.w += b.w;
  ((float4*)acc)[i] = a;
}

// ---------------------------------------------------------------------------
// Final gather + dot via V_WMMA_F32_16X16X4_F32.
// One wave handles 16 batch pairs: D = U(16x64) x I^T(64x16) accumulated over
// sixteen K=4 WMMA steps; diagonal D[j][j] = dot(u_j, i_j).
// f32 16x16x4 layout: lane L holds A[M=L&15][Kpair=(L>>4)*2 + {0,1}] and the
// B operand mirrors with N=L&15, so each lane gathers 8B from each row.
// ---------------------------------------------------------------------------
__global__ void lg_dot_wmma(const float* __restrict__ acc,
                            const int* __restrict__ uidx,
                            const int* __restrict__ iidx,
                            float* __restrict__ out) {
  int wave = blockIdx.x * (blockDim.x >> 5) + (threadIdx.x >> 5);
  int lane = threadIdx.x & 31;
  int b0 = wave * 16;
  if (b0 >= BATCH_B) return;          // wave-uniform; EXEC stays all-ones

  int row = lane & 15;
  int kh  = lane >> 4;                 // which K-pair of the 4-wide slab

  long u  = (long)uidx[b0 + row];
  long it = (long)NUSERS + (long)iidx[b0 + row];
  const float* ur = acc + u  * EMB_D + kh * 2;
  const float* ir = acc + it * EMB_D + kh * 2;

  v8f c = {0.f, 0.f, 0.f, 0.f, 0.f, 0.f, 0.f, 0.f};
#pragma unroll
  for (int k0 = 0; k0 < EMB_D; k0 += 4) {
    v2f a = *(const v2f*)(ur + k0);
    v2f b = *(const v2f*)(ir + k0);
    // 8 args: (neg_a, A, neg_b, B, c_mod, C, reuse_a, reuse_b)
    c = __builtin_amdgcn_wmma_f32_16x16x4_f32(false, a, false, b, (short)0, c,
                                              false, false);
  }

  // Diagonal extraction: lane L holds D[v + (L>>4)*8][L&15] in c[v].
  int v = (lane & 15) - kh * 8;
  if (v >= 0 && v < 8) {
    // both operands should carry the 1/(n_layers+1) mean -> 1/16 on the product
    out[b0 + (lane & 15)] = c[v] * (1.0f / ((NLAYER + 1) * (NLAYER + 1)));
  }
}

// ---------------------------------------------------------------------------
extern "C" void kernel_launch(void* const* d_in, const int* in_sizes, int n_in,
                              void* d_out, int out_size, void* d_ws, size_t ws_size,
                              hipStream_t stream) {
  const float* ue   = (const float*)d_in[0];
  const float* ie   = (const float*)d_in[1];
  const int*   rows = (const int*)d_in[2];
  const int*   cols = (const int*)d_in[3];
  const float* vals = (const float*)d_in[4];
  const int*   uix  = (const int*)d_in[5];
  const int*   iix  = (const int*)d_in[6];
  float*       out  = (float*)d_out;

  float* acc = (float*)d_ws;
  float* eA  = acc + (size_t)NTOT * EMB_D;
  float* eB  = eA + (size_t)NTOT * EMB_D;

  const long nT4 = (long)NTOT * EMB_D / 4;
  const int  blk = 256;
  const int  gT4 = (int)((nT4 + blk - 1) / blk);

  lg_init<<<gT4, blk, 0, stream>>>(ue, ie, acc, eA);

  float* ein = eA;
  float* eout = eB;
  for (int l = 0; l < NLAYER; ++l) {
    lg_zero<<<gT4, blk, 0, stream>>>(eout);
    long nthr = (long)NNZ_E * 16;
    int  gE   = (int)((nthr + blk - 1) / blk);
    lg_spmm<<<gE, blk, 0, stream>>>(rows, cols, vals, ein, eout);
    lg_accum<<<gT4, blk, 0, stream>>>(acc, eout);
    float* t = ein; ein = eout; eout = t;
  }

  // 1024 tiles of 16 pairs; 4 waves per block -> 256 blocks of 128 threads
  lg_dot_wmma<<<BATCH_B / 16 / 4, 128, 0, stream>>>(acc, uix, iix, out);
}